// ChamferDistanceLoss_28724741276335
// MI455X (gfx1250) — compile-verified
//
#include <hip/hip_runtime.h>

// CDNA5 (gfx1250) wave32 WMMA types
typedef __attribute__((ext_vector_type(2))) float v2f;  // A/B for f32 16x16x4: 64 vals / 32 lanes
typedef __attribute__((ext_vector_type(8))) float v8f;  // C/D 16x16 f32: 256 vals / 32 lanes

#define NPTS  8192
#define BATCH 4

__global__ void chamfer_zero_ws(float* ws) { ws[0] = 0.0f; }

// Computes  sum_{b,n} [ min_m( -2*q_n . t_m + ||t_m||^2 ) + ||q_n||^2 ]
// and atomically accumulates into accum[0].  Called twice (roles swapped).
//
// Grid: (NPTS/128, BATCH), block 256 (= 8 wave32), dynamic LDS = NPTS*4 floats.
__global__ __launch_bounds__(256)
void chamfer_dir_kernel(const float* __restrict__ query,
                        const float* __restrict__ db,
                        float* __restrict__ accum) {
    extern __shared__ float sdb[];          // [NPTS][4] = {x, y, z, ||t||^2}: 128 KB
    const int b   = blockIdx.y;
    const int tid = threadIdx.x;

    // Stage target cloud into LDS with the augmented norm component precomputed
    // once per block (instead of per wave per tile).  Stride-4 records give an
    // aligned ds_load_b64 per lane in the hot loop.
    {
        const float* src = db + (size_t)b * NPTS * 3;
        float4*      dst = (float4*)sdb;
        for (int i = tid; i < NPTS; i += 256) {
            const float x = src[i * 3 + 0];
            const float y = src[i * 3 + 1];
            const float z = src[i * 3 + 2];
            dst[i] = make_float4(x, y, z, x * x + y * y + z * z);
        }
    }
    __syncthreads();

    const int wave = tid >> 5;
    const int lane = tid & 31;
    const int half = lane >> 4;       // 0: lanes 0-15, 1: lanes 16-31
    const int l15  = lane & 15;
    const int n0   = (blockIdx.x * 8 + wave) * 16;   // this wave's 16-query tile

    // Load query point for row (n0 + l15); both lane halves load the same point.
    const float* qp = query + ((size_t)b * NPTS + (size_t)(n0 + l15)) * 3;
    const float qx = qp[0], qy = qp[1], qz = qp[2];

    // A (16x4 f32, ISA layout): VGPR0 = K0 (lanes 0-15) / K2 (lanes 16-31),
    //                           VGPR1 = K1 / K3.  Augmented row = [-2x,-2y,-2z, 1].
    v2f a;
    a.x = half ? (-2.0f * qz) : (-2.0f * qx);
    a.y = half ?        1.0f  : (-2.0f * qy);

    float mins[8];
#pragma unroll
    for (int r = 0; r < 8; ++r) mins[r] = __builtin_inff();

    // B (4x16 f32): rows striped across lanes, K split across VGPR/half.
    // Augmented col = [x, y, z, ||t||^2].  Lanes 0-15 need (K0,K1) = (x,y);
    // lanes 16-31 need (K2,K3) = (z,||t||^2) -> one aligned 8-byte LDS load,
    // no operand fix-up VALU.
    const float* tilebase = sdb + ((l15 << 2) + (half << 1));

    // Software-pipelined scan over all targets in 16-wide tiles: prefetch tile
    // t+1 before consuming tile t so the ds_load -> v_wmma wait is covered by a
    // full loop group instead of a handful of VALU ops.  The wrap-around index
    // keeps the epilogue branchless (EXEC must stay all-ones around WMMA).
    float2 cur = *(const float2*)(tilebase);
#pragma unroll 8
    for (int m0 = 0; m0 < NPTS; m0 += 16) {
        const int mn = (m0 + 16) & (NPTS - 1);          // next tile (wraps once)
        const float2 nxt = *(const float2*)(tilebase + (mn << 2));

        v2f bv;
        bv.x = cur.x;
        bv.y = cur.y;

        v8f c = {};
        c = __builtin_amdgcn_wmma_f32_16x16x4_f32(
                /*neg_a=*/false, a, /*neg_b=*/false, bv,
                /*c_mod=*/(short)0, c, /*reuse_a=*/false, /*reuse_b=*/false);

        // C/D layout: VGPR r holds row (r + 8*half), column = l15.
#pragma unroll
        for (int r = 0; r < 8; ++r) mins[r] = fminf(mins[r], c[r]);

        cur = nxt;
    }

    // Min-reduce each row across its 16-lane group (xor masks < 16 stay in-half).
    float rowsum = 0.0f;
#pragma unroll
    for (int r = 0; r < 8; ++r) {
        float v = mins[r];
        v = fminf(v, __shfl_xor(v, 1, 32));
        v = fminf(v, __shfl_xor(v, 2, 32));
        v = fminf(v, __shfl_xor(v, 4, 32));
        v = fminf(v, __shfl_xor(v, 8, 32));
        if (l15 == 0) rowsum += v;    // lane 0: rows 0..7, lane 16: rows 8..15
    }

    // ||q||^2 contribution: each of the 16 query points counted once (lanes 0-15).
    float lanesum = rowsum + ((half == 0) ? (qx * qx + qy * qy + qz * qz) : 0.0f);

    // Full-wave sum, one atomic per wave.
    lanesum += __shfl_xor(lanesum, 16, 32);
    lanesum += __shfl_xor(lanesum,  8, 32);
    lanesum += __shfl_xor(lanesum,  4, 32);
    lanesum += __shfl_xor(lanesum,  2, 32);
    lanesum += __shfl_xor(lanesum,  1, 32);
    if (lane == 0) atomicAdd(accum, lanesum);
}

__global__ void chamfer_finalize(const float* __restrict__ accum,
                                 float* __restrict__ out) {
    // x_near.mean() + y_near.mean(); both means are over BATCH*NPTS elements.
    out[0] = accum[0] * (1.0f / (float)(BATCH * NPTS));
}

extern "C" void kernel_launch(void* const* d_in, const int* in_sizes, int n_in,
                              void* d_out, int out_size, void* d_ws, size_t ws_size,
                              hipStream_t stream) {
    const float* predict = (const float*)d_in[0];   // [4, 8192, 3] f32
    const float* target  = (const float*)d_in[1];   // [4, 8192, 3] f32
    float* out = (float*)d_out;                     // scalar f32
    float* acc = (float*)d_ws;                      // 1 f32 accumulator

    chamfer_zero_ws<<<1, 1, 0, stream>>>(acc);

    const dim3 grid(NPTS / 128, BATCH);             // 8 waves/block * 16 rows/wave
    const size_t lds_bytes = (size_t)NPTS * 4 * sizeof(float);   // 128 KB < 320 KB/WGP

    chamfer_dir_kernel<<<grid, 256, lds_bytes, stream>>>(predict, target, acc);
    chamfer_dir_kernel<<<grid, 256, lds_bytes, stream>>>(target, predict, acc);

    chamfer_finalize<<<1, 1, 0, stream>>>(acc, out);
}